// SharedWriter_35270271435251
// MI455X (gfx1250) — compile-verified
//
#include <hip/hip_runtime.h>

typedef __attribute__((ext_vector_type(2))) float v2f;
typedef __attribute__((ext_vector_type(4))) float v4f;
typedef __attribute__((ext_vector_type(8))) float v8f;

#define HIDDEN 1024
#define Bn     32
#define Tn     2048
#define TS     2045   // T-3 steps actually scanned
#define FASTN  64
#define SLOWN  192
// sigmoid(x) >= 0.4  <=>  x >= ln(0.4/0.6)
#define WTHR (-0.40546510810816438f)

// LDS weight buffer layout (floats), chosen for bank-conflict-free ds_load_b64:
//   [0, 1024)      Wg
//   [1032, 2056)   Wd          (start%64 == 8 -> banks offset +8 from Wg)
//   [2064, 3136)   zeros       (start%64 == 16; inactive row r reads 2064+2r+kb)
#define WGOFF 0
#define WDOFF 1032
#define ZOFF  2064
#define WBUFN 3136

// ---------------------------------------------------------------------------
// Phase 1: gdot[b][t] = h[b,t,:]·Wg ; ddot[b][t] = h[b,t,:]·Wd
// One wave per 16-token tile. WMMA f32 16x16x4:
//   A (16x4)  = h rows (tokens on M): lanes 0-15 {K=0,K=1}, lanes 16-31 {K=2,K=3}.
//   B (4x16)  = weight columns (N=0 -> Wg, N=1 -> Wd, rest read zeros from LDS).
//   D (16x16) : lane0: Wg-dots tokens 0..7, lane16: 8..15; lanes 1/17: Wd-dots.
// ---------------------------------------------------------------------------
__global__ __launch_bounds__(128) void p1_proj(const float* __restrict__ h,
                                               const float* __restrict__ Wg,
                                               const float* __restrict__ Wd,
                                               float* __restrict__ gdot,
                                               float* __restrict__ ddot) {
  __shared__ __align__(16) float wbuf[WBUFN];
  for (int i = threadIdx.x; i < WBUFN; i += 128) {
    float v = 0.f;
    if (i < HIDDEN) v = Wg[i];
    else if (i >= WDOFF && i < WDOFF + HIDDEN) v = Wd[i - WDOFF];
    wbuf[i] = v;
  }
  __syncthreads();

  const int wave = threadIdx.x >> 5;
  const int lane = threadIdx.x & 31;
  const int tile = blockIdx.x * 4 + wave;     // 4096 tiles = 32 batches * 128
  const int b    = tile >> 7;
  const int t0   = (tile & 127) << 4;         // 16 tokens/tile (t up to 2047 is
  const int half = lane >> 4;                 // allocated; extras ignored later)
  const int row  = lane & 15;

  const float* hbase = h + ((size_t)(b * Tn + t0 + row)) * HIDDEN + 2 * half;

  // Branchless B operand base: rows 0/1 -> real weights, rows 2..15 -> zeros.
  int belem;
  if (row == 0)      belem = WGOFF + 2 * half;
  else if (row == 1) belem = WDOFF + 2 * half;
  else               belem = ZOFF + 2 * row;  // same addr both halves -> broadcast
  const float* wbase = wbuf + belem;

  v8f acc = {0.f, 0.f, 0.f, 0.f, 0.f, 0.f, 0.f, 0.f};
  for (int kb = 0; kb < HIDDEN; kb += 32) {
    v2f a[8], w[8];
#pragma unroll
    for (int j = 0; j < 8; ++j) {
      a[j] = *(const v2f*)(hbase + kb + 4 * j);   // global_load_b64, imm offsets
      w[j] = *(const v2f*)(wbase + kb + 4 * j);   // ds_load_b64, conflict-free
    }
#pragma unroll
    for (int j = 0; j < 8; ++j)
      acc = __builtin_amdgcn_wmma_f32_16x16x4_f32(false, a[j], false, w[j],
                                                  (short)0, acc, false, false);
  }

  if (row < 2) {
    float* outp = ((row & 1) ? ddot : gdot) + b * Tn + t0 + 8 * half;
    v4f* ap = (v4f*)&acc;
    *(v4f*)(outp)     = ap[0];
    *(v4f*)(outp + 4) = ap[1];
  }
}

// ---------------------------------------------------------------------------
// Phase 2: sequential scan. One wave per batch. Lane l owns fast slots {l, l+32}
// (ds value + source-token index) and slow slots {l+32j} (source-token index).
// Fast fills 0..63 sequentially (fu never clears); afterwards target = argmin(ds)
// with first-index tie rule (ballot + ffs). Slow selection is round-robin.
// ---------------------------------------------------------------------------
__global__ __launch_bounds__(32) void p2_scan(const float* __restrict__ gdot,
                                              const float* __restrict__ ddot,
                                              const float* __restrict__ bgp,
                                              const float* __restrict__ bdp,
                                              int* __restrict__ ffrom,
                                              int* __restrict__ sfrom,
                                              float* __restrict__ out_fu,
                                              float* __restrict__ out_su) {
  const int b    = blockIdx.x;
  const int lane = threadIdx.x;
  const float bg = bgp[0], bd = bdp[0];
  const float* gp = gdot + b * Tn;
  const float* dp = ddot + b * Tn;

  float ds0 = 3.0e38f, ds1 = 3.0e38f;       // never consulted before fast is full
  int   f0 = -1, f1 = -1;
  int   s0 = -1, s1 = -1, s2 = -1, s3 = -1, s4 = -1, s5 = -1;
  int   nfast = 0, ndem = 0;

  for (int tc = 0; tc < TS; tc += 32) {
    float gt = gp[tc + lane];                // prefetch chunk (t<=2047 valid)
    float dt = dp[tc + lane];
    int kmax = (TS - tc < 32) ? (TS - tc) : 32;
    for (int k = 0; k < kmax; ++k) {
      float g = __shfl(gt, k, 32);
      if (g + bg < WTHR) continue;           // no write, no observable state change
      float dv = __shfl(dt, k, 32) + bd;
      int   t  = tc + k;
      int   slot;
      if (nfast < FASTN) {
        slot = nfast++;                      // first free fast slot
      } else {
        // argmin over ds[0..63], first index on ties
        float m = fminf(ds0, ds1);
        for (int off = 16; off > 0; off >>= 1) m = fminf(m, __shfl_xor(m, off, 32));
        unsigned bal0 = (unsigned)__ballot(ds0 == m);
        unsigned bal1 = (unsigned)__ballot(ds1 == m);
        int dem  = bal0 ? (__ffs(bal0) - 1) : (32 + __ffs(bal1) - 1);
        int dsel = (dem < 32) ? f0 : f1;
        int dtok = __shfl(dsel, dem & 31, 32);   // demoted token (pre-overwrite)
        int ss = ndem % SLOWN;               // slow slot: sequential fill then FIFO
        ndem++;
        int j = ss >> 5, ol = ss & 31;
        if (lane == ol) {
          if      (j == 0) s0 = dtok; else if (j == 1) s1 = dtok;
          else if (j == 2) s2 = dtok; else if (j == 3) s3 = dtok;
          else if (j == 4) s4 = dtok; else             s5 = dtok;
        }
        slot = dem;
      }
      if (slot < 32) { if (lane == slot)      { ds0 = dv; f0 = t; } }
      else           { if (lane == slot - 32) { ds1 = dv; f1 = t; } }
    }
  }

  int* ff = ffrom + b * FASTN;
  ff[lane] = f0; ff[lane + 32] = f1;
  int* sf = sfrom + b * SLOWN;
  sf[lane] = s0; sf[lane + 32] = s1; sf[lane + 64]  = s2;
  sf[lane + 96] = s3; sf[lane + 128] = s4; sf[lane + 160] = s5;

  float* fu = out_fu + b * FASTN;
  fu[lane]      = (lane < nfast) ? 1.f : 0.f;
  fu[lane + 32] = (lane + 32 < nfast) ? 1.f : 0.f;
  int nsu = (ndem < SLOWN) ? ndem : SLOWN;
  float* su = out_su + b * SLOWN;
#pragma unroll
  for (int j = 0; j < 6; ++j)
    su[lane + 32 * j] = (lane + 32 * j < nsu) ? 1.f : 0.f;
}

// ---------------------------------------------------------------------------
// Phase 3: gather fm/sm rows from h by recorded token index (zeros if unused).
// One 256-thread block per 1024-float row; 16B vector copies.
// ---------------------------------------------------------------------------
__global__ __launch_bounds__(256) void p3_gather(const float* __restrict__ h,
                                                 const int* __restrict__ ffrom,
                                                 const int* __restrict__ sfrom,
                                                 float* __restrict__ out) {
  const int rid = blockIdx.x;
  int b, srcIdx;
  float* dst;
  if (rid < Bn * FASTN) {
    b = rid / FASTN;
    srcIdx = ffrom[rid];
    dst = out + (size_t)rid * HIDDEN;
  } else {
    int r = rid - Bn * FASTN;
    b = r / SLOWN;
    srcIdx = sfrom[r];
    dst = out + (size_t)Bn * FASTN * HIDDEN + (size_t)r * HIDDEN;
  }
  const int i = threadIdx.x * 4;
  v4f v = {0.f, 0.f, 0.f, 0.f};
  if (srcIdx >= 0) v = *(const v4f*)(h + ((size_t)(b * Tn + srcIdx)) * HIDDEN + i);
  *(v4f*)(dst + i) = v;
}

// ---------------------------------------------------------------------------
// d_ws layout (32-bit units): [0,65536) gdot[B][2048]; [65536,131072) ddot;
// [131072,133120) ffrom int[B][64]; [133120,139264) sfrom int[B][192].
// Requires ws_size >= 557,056 bytes.
// ---------------------------------------------------------------------------
extern "C" void kernel_launch(void* const* d_in, const int* in_sizes, int n_in,
                              void* d_out, int out_size, void* d_ws, size_t ws_size,
                              hipStream_t stream) {
  const float* h  = (const float*)d_in[0];
  const float* Wg = (const float*)d_in[1];
  const float* bg = (const float*)d_in[2];
  const float* Wd = (const float*)d_in[3];
  const float* bd = (const float*)d_in[4];
  float* out = (float*)d_out;

  float* gdot = (float*)d_ws;
  float* ddot = gdot + Bn * Tn;
  int*   ffrom = (int*)(ddot + Bn * Tn);
  int*   sfrom = ffrom + Bn * FASTN;

  float* out_fu = out + (size_t)Bn * FASTN * HIDDEN + (size_t)Bn * SLOWN * HIDDEN;
  float* out_su = out_fu + Bn * FASTN;

  p1_proj<<<1024, 128, 0, stream>>>(h, Wg, Wd, gdot, ddot);
  p2_scan<<<Bn, 32, 0, stream>>>(gdot, ddot, bg, bd, ffrom, sfrom, out_fu, out_su);
  p3_gather<<<Bn * (FASTN + SLOWN), 256, 0, stream>>>(h, ffrom, sfrom, out);
}